// MSAColumnGlobalAttention_ckpt_33681133535497
// MI455X (gfx1250) — compile-verified
//
#include <hip/hip_runtime.h>
#include <hip/hip_bf16.h>
#include <math.h>

typedef float v2f __attribute__((ext_vector_type(2)));
typedef float v8f __attribute__((ext_vector_type(8)));
typedef unsigned int v4u __attribute__((ext_vector_type(4)));
typedef int v4i __attribute__((ext_vector_type(4)));
typedef int v8i __attribute__((ext_vector_type(8)));

#define S_DIM 4096
#define I_DIM 1024
#define C_DIM 8
#define NTHREADS 1024
#define NWAVES 32
#define ROWS_PER_THREAD (S_DIM / NTHREADS)

// LDS layout (floats)
#define OFF_MN   0        // 4096 x 8  (raw column tile, then normalized in-place)
#define OFF_KB   32768    // 4096
#define OFF_MB   36864    // 4096 (mask column)
#define OFF_RED  40960    // 32 x 16
#define OFF_TOT  41472    // 16
#define OFF_OV   41488    // 8
#define SMEM_FLOATS 41496

#if defined(__has_builtin)
#if __has_builtin(__builtin_amdgcn_tensor_load_to_lds) && __has_builtin(__builtin_amdgcn_s_wait_tensorcnt)
#define USE_TDM 1
#endif
#endif

__device__ __forceinline__ float wave_sum(float v) {
#pragma unroll
  for (int o = 16; o > 0; o >>= 1) v += __shfl_down(v, o, 32);
  return v;
}
__device__ __forceinline__ float wave_max(float v) {
#pragma unroll
  for (int o = 16; o > 0; o >>= 1) v = fmaxf(v, __shfl_down(v, o, 32));
  return v;
}

// Build B-operand (4x16, K-major in VGPRs, K-halves split across lane groups)
// for D(s,N) = sum_K A(s,K) * B(K,N) with B[K][N] = w[N*8+K] (w is 8x8 row-major).
__device__ __forceinline__ v2f gather_b(const float* __restrict__ w, int lane, int blk) {
  int N = lane & 15;
  int k = ((lane >> 4) << 1) + (blk << 2);
  v2f b;
  if (N < 8) { b.x = w[N * 8 + k]; b.y = w[N * 8 + k + 1]; }
  else       { b.x = 0.f;          b.y = 0.f; }
  return b;
}

#ifdef USE_TDM
// Issue a 2D TDM column-gather: tile_dim0 elems/row (f32), nrows rows,
// row stride `stride` (f32 units), from global `gaddr` into LDS byte addr `lds_byte`.
__device__ __forceinline__ void tdm_load_2d(unsigned long long gaddr, unsigned lds_byte,
                                            unsigned dim0, unsigned nrows, unsigned stride) {
  v4u g0;
  g0.x = 1u;                                    // count=1 (valid), user mode, no gather
  g0.y = lds_byte;                              // lds_addr [63:32]
  g0.z = (unsigned)(gaddr & 0xFFFFFFFFull);     // global_addr [95:64]
  g0.w = (unsigned)((gaddr >> 32) & 0x01FFFFFFull) | 0x80000000u; // addr[56:32] | type=2
  v8i g1;
  g1[0] = (int)(2u << 16);                      // workgroup_mask=0, data_size=2 (4B)
  g1[1] = (int)(dim0 << 16);                    // tensor_dim0 low16 at bits[63:48]
  g1[2] = (int)((dim0 >> 16) | (nrows << 16));  // tensor_dim0 hi | tensor_dim1 low16
  g1[3] = (int)((nrows >> 16) | (dim0 << 16));  // tensor_dim1 hi | tile_dim0
  g1[4] = (int)nrows;                           // tile_dim1 | tile_dim2=0
  g1[5] = (int)stride;                          // tensor_dim0_stride low32
  g1[6] = 0;                                    // stride0 hi16 | stride1 low16
  g1[7] = 0;
  v4i z4 = {0, 0, 0, 0};
#if __clang_major__ >= 23
  v8i z8 = {0, 0, 0, 0, 0, 0, 0, 0};
  __builtin_amdgcn_tensor_load_to_lds(g0, g1, z4, z4, z8, 0);
#else
  __builtin_amdgcn_tensor_load_to_lds(g0, g1, z4, z4, 0);
#endif
}
#endif

__global__ __launch_bounds__(NTHREADS)
void msa_col_global_attn(const float* __restrict__ m, const float* __restrict__ mask,
                         const float* __restrict__ ln_w, const float* __restrict__ ln_b,
                         const float* __restrict__ wq, const float* __restrict__ wk,
                         const float* __restrict__ wv, const float* __restrict__ wg,
                         const float* __restrict__ bg, const float* __restrict__ wo,
                         const float* __restrict__ bo, float* __restrict__ out) {
  extern __shared__ float smem[];
  float* mn  = smem + OFF_MN;
  float* kb  = smem + OFF_KB;
  float* mb  = smem + OFF_MB;
  float* red = smem + OFF_RED;
  float* tot = smem + OFF_TOT;
  float* ov  = smem + OFF_OV;
  float* stage = kb; // reused in phase 3 (32 waves * 128 floats)

  const int i    = blockIdx.x;
  const int tid  = threadIdx.x;
  const int lane = tid & 31;
  const int w    = tid >> 5;

  float lw[8], lbv[8], wkl[8], wvl[8];
#pragma unroll
  for (int j = 0; j < 8; j++) { lw[j] = ln_w[j]; lbv[j] = ln_b[j]; wkl[j] = wk[j]; wvl[j] = wv[j]; }

  // ------------- Phase 0: DMA the whole (4096 x 8) column + mask column into LDS -------------
#ifdef USE_TDM
  if (w == 0) { // one wave issues both DMAs (TDM ignores EXEC; per-wave TENSORcnt)
    tdm_load_2d((unsigned long long)(uintptr_t)(m + (size_t)i * C_DIM),
                OFF_MN * 4u, C_DIM, S_DIM, I_DIM * C_DIM);
    tdm_load_2d((unsigned long long)(uintptr_t)(mask + (size_t)i),
                OFF_MB * 4u, 1u, S_DIM, I_DIM);
    __builtin_amdgcn_s_wait_tensorcnt(0);
  }
  __syncthreads();
#endif

  // ---------------- Phase 1: LayerNorm (in place), k, masked q-pool accumulation ---------------
  float qacc[8] = {0.f, 0.f, 0.f, 0.f, 0.f, 0.f, 0.f, 0.f};
  float macc = 0.f;
#pragma unroll
  for (int r = 0; r < ROWS_PER_THREAD; r++) {
    const int s = r * NTHREADS + tid;
    float x[8];
    float msk;
#ifdef USE_TDM
    float4 x0 = ((const float4*)(mn + (size_t)s * 8))[0];
    float4 x1 = ((const float4*)(mn + (size_t)s * 8))[1];
    msk = mb[s];
#else
    const float* rowp = m + (size_t)s * (I_DIM * C_DIM) + (size_t)i * C_DIM;
    if (r + 1 < ROWS_PER_THREAD)
      __builtin_prefetch(m + (size_t)(s + NTHREADS) * (I_DIM * C_DIM) + (size_t)i * C_DIM, 0, 1);
    float4 x0 = ((const float4*)rowp)[0];
    float4 x1 = ((const float4*)rowp)[1];
    msk = mask[(size_t)s * I_DIM + i];
    mb[s] = msk;
#endif
    x[0]=x0.x; x[1]=x0.y; x[2]=x0.z; x[3]=x0.w; x[4]=x1.x; x[5]=x1.y; x[6]=x1.z; x[7]=x1.w;
    float mu = 0.f;
#pragma unroll
    for (int j = 0; j < 8; j++) mu += x[j];
    mu *= 0.125f;
    float var = 0.f;
#pragma unroll
    for (int j = 0; j < 8; j++) { float d = x[j] - mu; var += d * d; }
    var *= 0.125f;
    const float inv = rsqrtf(var + 1e-5f);
    float row[8];
    float ks = 0.f;
#pragma unroll
    for (int j = 0; j < 8; j++) {
      row[j] = (x[j] - mu) * inv * lw[j] + lbv[j];
      ks += row[j] * wkl[j];
      qacc[j] += row[j] * msk;
    }
    macc += msk;
    float4* dst = (float4*)(mn + (size_t)s * 8);
    dst[0] = make_float4(row[0], row[1], row[2], row[3]);
    dst[1] = make_float4(row[4], row[5], row[6], row[7]);
    kb[s] = ks;
  }
  // block reduce: 8 sums + 1 count
#pragma unroll
  for (int j = 0; j < 8; j++) { float v = wave_sum(qacc[j]); if (lane == 0) red[w * 16 + j] = v; }
  { float v = wave_sum(macc); if (lane == 0) red[w * 16 + 8] = v; }
  __syncthreads();
  if (tid < 9) {
    float v = 0.f;
    for (int ww = 0; ww < NWAVES; ww++) v += red[ww * 16 + tid];
    tot[tid] = v;
  }
  __syncthreads();

  // q = (pooled @ wq^T) * c_h^-0.5 (c_h=1 -> scale 1)
  const float minv = 1.f / (tot[8] + 1e-5f);
  float pooled[8];
#pragma unroll
  for (int c = 0; c < 8; c++) pooled[c] = tot[c] * minv;
  float qh[8];
#pragma unroll
  for (int h = 0; h < 8; h++) {
    float a = 0.f;
#pragma unroll
    for (int c = 0; c < 8; c++) a += pooled[c] * wq[h * 8 + c];
    qh[h] = a;
  }

  // ---------------- Phase 2: softmax over s per head, o_h = sum softmax*v ----------------
  float lmax[8];
#pragma unroll
  for (int h = 0; h < 8; h++) lmax[h] = -3.4e38f;
#pragma unroll
  for (int r = 0; r < ROWS_PER_THREAD; r++) {
    const int s = r * NTHREADS + tid;
    const float ks = kb[s];
    const float bs = 1e9f * (mb[s] - 1.f);
#pragma unroll
    for (int h = 0; h < 8; h++) lmax[h] = fmaxf(lmax[h], qh[h] * ks + bs);
  }
#pragma unroll
  for (int h = 0; h < 8; h++) { float v = wave_max(lmax[h]); if (lane == 0) red[w * 16 + h] = v; }
  __syncthreads();
  if (tid < 8) {
    float v = -3.4e38f;
    for (int ww = 0; ww < NWAVES; ww++) v = fmaxf(v, red[ww * 16 + tid]);
    tot[tid] = v;
  }
  __syncthreads();
  float mxh[8];
#pragma unroll
  for (int h = 0; h < 8; h++) mxh[h] = tot[h];

  float esum[8] = {0.f, 0.f, 0.f, 0.f, 0.f, 0.f, 0.f, 0.f};
  float vacc[8] = {0.f, 0.f, 0.f, 0.f, 0.f, 0.f, 0.f, 0.f};
#pragma unroll
  for (int r = 0; r < ROWS_PER_THREAD; r++) {
    const int s = r * NTHREADS + tid;
    const float ks = kb[s];
    const float bs = 1e9f * (mb[s] - 1.f);
    const float* rowv = mn + (size_t)s * 8;
    float vs = 0.f;
#pragma unroll
    for (int c = 0; c < 8; c++) vs += rowv[c] * wvl[c];
#pragma unroll
    for (int h = 0; h < 8; h++) {
      float p = __expf(qh[h] * ks + bs - mxh[h]);
      esum[h] += p;
      vacc[h] += p * vs;
    }
  }
#pragma unroll
  for (int h = 0; h < 8; h++) { float v = wave_sum(esum[h]); if (lane == 0) red[w * 16 + h] = v; }
#pragma unroll
  for (int h = 0; h < 8; h++) { float v = wave_sum(vacc[h]); if (lane == 0) red[w * 16 + 8 + h] = v; }
  __syncthreads();
  if (tid < 16) {
    float v = 0.f;
    for (int ww = 0; ww < NWAVES; ww++) v += red[ww * 16 + tid];
    tot[tid] = v;
  }
  __syncthreads();
  if (tid < 8) ov[tid] = tot[8 + tid] / tot[tid];
  __syncthreads();

  // ---------------- Phase 3: gate + output projection via WMMA f32 16x16x4 ----------------
  // g = sigmoid(mn @ wg^T + bg); t = o * g; out = t @ wo^T + bo
  const v2f B1lo = gather_b(wg, lane, 0), B1hi = gather_b(wg, lane, 1);
  const v2f B2lo = gather_b(wo, lane, 0), B2hi = gather_b(wo, lane, 1);
  const int Nl = lane & 15;
  const int kh = (lane >> 4) << 1;         // K offset for A-operand (0 or 2)
  const int M0 = (lane >> 4) << 3;         // D-layout row offset (0 or 8)
  const float bgN = (Nl < 8) ? bg[Nl] : 0.f;
  const float boN = (Nl < 8) ? bo[Nl] : 0.f;
  const float oN  = (Nl < 8) ? ov[Nl] : 0.f;
  float* st = stage + w * 128;             // per-wave 16x8 staging tile
  const int srow = lane >> 1, shalf = lane & 1;

  for (int tile = 0; tile < (S_DIM / NWAVES) / 16; tile++) {
    const int s0 = w * (S_DIM / NWAVES) + tile * 16;
    v2f A1lo = *(const v2f*)(mn + (size_t)(s0 + Nl) * 8 + kh);
    v2f A1hi = *(const v2f*)(mn + (size_t)(s0 + Nl) * 8 + kh + 4);
    v8f acc = {0.f, 0.f, 0.f, 0.f, 0.f, 0.f, 0.f, 0.f};
    acc = __builtin_amdgcn_wmma_f32_16x16x4_f32(false, A1lo, false, B1lo, (short)0, acc, false, false);
    acc = __builtin_amdgcn_wmma_f32_16x16x4_f32(false, A1hi, false, B1hi, (short)0, acc, false, false);
    // gate: t = o_h * sigmoid(x + bg) -- fast reciprocal instead of IEEE divide
    float tv[8];
#pragma unroll
    for (int r = 0; r < 8; r++)
      tv[r] = oN * __builtin_amdgcn_rcpf(1.f + __expf(-(acc[r] + bgN)));
    // D-layout -> A-layout via per-wave LDS stage (DS ops of one wave are in-order)
    if (Nl < 8) {
#pragma unroll
      for (int r = 0; r < 8; r++) st[(M0 + r) * 8 + Nl] = tv[r];
    }
    v2f A2lo = *(const v2f*)(st + (size_t)Nl * 8 + kh);
    v2f A2hi = *(const v2f*)(st + (size_t)Nl * 8 + kh + 4);
    v8f acc2 = {0.f, 0.f, 0.f, 0.f, 0.f, 0.f, 0.f, 0.f};
    acc2 = __builtin_amdgcn_wmma_f32_16x16x4_f32(false, A2lo, false, B2lo, (short)0, acc2, false, false);
    acc2 = __builtin_amdgcn_wmma_f32_16x16x4_f32(false, A2hi, false, B2hi, (short)0, acc2, false, false);
    // stage D2 tile, then one coalesced b128 store per lane (all 32 lanes active)
    if (Nl < 8) {
#pragma unroll
      for (int r = 0; r < 8; r++) st[(M0 + r) * 8 + Nl] = acc2[r] + boN;
    }
    float4 o4 = ((const float4*)st)[lane];
    *(float4*)(out + (size_t)(s0 + srow) * (I_DIM * C_DIM) + (size_t)i * C_DIM + shalf * 4) = o4;
  }
}

extern "C" void kernel_launch(void* const* d_in, const int* in_sizes, int n_in,
                              void* d_out, int out_size, void* d_ws, size_t ws_size,
                              hipStream_t stream) {
  const float* m    = (const float*)d_in[0];
  const float* mask = (const float*)d_in[1];
  const float* ln_w = (const float*)d_in[2];
  const float* ln_b = (const float*)d_in[3];
  const float* wq   = (const float*)d_in[4];
  const float* wk   = (const float*)d_in[5];
  const float* wv   = (const float*)d_in[6];
  const float* wg   = (const float*)d_in[7];
  const float* bg   = (const float*)d_in[8];
  const float* wo   = (const float*)d_in[9];
  const float* bo   = (const float*)d_in[10];
  float* out = (float*)d_out;

  const size_t smem_bytes = (size_t)SMEM_FLOATS * sizeof(float); // ~162 KB (<= 320 KB/WG on CDNA5)
  (void)hipFuncSetAttribute((const void*)msa_col_global_attn,
                            hipFuncAttributeMaxDynamicSharedMemorySize, (int)smem_bytes);
  msa_col_global_attn<<<I_DIM, NTHREADS, smem_bytes, stream>>>(
      m, mask, ln_w, ln_b, wq, wk, wv, wg, bg, wo, bo, out);
}